// Hadamard_9354438771401
// MI455X (gfx1250) — compile-verified
//
#include <hip/hip_runtime.h>

// ---------------------------------------------------------------------------
// Hadamard (12-qubit, N=4096) state-vector transform for MI455X / gfx1250.
//
// U = H16 (x) H16 (x) H16. Each 4096-length row, viewed as a 16x16x16 tensor,
// gets a 16x16 Hadamard contraction on each of the 3 modes; each mode is 16
// independent 16x16(x16) tiles executed as 4 chained V_WMMA_F32_16X16X4_F32
// ops (fp32 A/B/C matches the complex64 reference; the unitary is real so
// re/im transform independently). ~85x fewer FLOPs than the naive GEMM.
//
// Per workgroup (256 threads = 8 waves, wave32): one row (re+im = 32 KB LDS),
// async global->LDS streaming fill (GLOBAL_LOAD_ASYNC_TO_LDS_B128, ASYNCcnt-
// tracked, bypasses VGPRs), 3 in-place WMMA stages separated by single
// barriers, interleaved complex64 NT store.
// Roofline: 128 MiB in + 128 MiB out => ~11.5 us floor at 23.3 TB/s; total
// WMMA work (1.57M f32 wmma ops) is far below that on any plausible rate.
// ---------------------------------------------------------------------------

typedef float v2f __attribute__((ext_vector_type(2)));
typedef float v4f __attribute__((ext_vector_type(4)));
typedef float v8f __attribute__((ext_vector_type(8)));
typedef int   v4i __attribute__((ext_vector_type(4)));

#define NSTATE 4096   // 2^12 state dimension

// Async global->LDS path (CDNA5): guarded so either toolchain compiles.
#if defined(__HIP_DEVICE_COMPILE__) && \
    __has_builtin(__builtin_amdgcn_global_load_async_to_lds_b128)
#define USE_ASYNC_LDS 1
#else
#define USE_ASYNC_LDS 0
#endif

static __device__ __forceinline__ v8f wmma_f32_16x16x4(v2f a, v2f b, v8f c) {
  // 8-arg VOP3P form: (neg_a, A, neg_b, B, c_mod, C, reuse_a, reuse_b)
  return __builtin_amdgcn_wmma_f32_16x16x4_f32(
      false, a, false, b, (short)0, c, false, false);
}

// H16[a][k] * 0.25 built in integer ALU: 0.25f with sign = parity(a & k).
// No VCC compares/cndmask -> co-executes freely with WMMA.
static __device__ __forceinline__ float hsign(int a, int k) {
  unsigned s = ((unsigned)__popc(a & k)) << 31;   // parity bit -> sign bit
  return __int_as_float(0x3E800000u ^ s);         // +/- 0.25f
}

__global__ __launch_bounds__(256, 4) void hadamard4096_fwht_wmma(
    const float* __restrict__ xre,
    const float* __restrict__ xim,
    float* __restrict__ out)
{
  // [component][element]; 32 KB of the 320 KB/WGP LDS -> many WGs per WGP.
  __shared__ float lds[2][NSTATE];

  const int row  = blockIdx.x;        // 0..4095  (flattened B*M)
  const int tid  = threadIdx.x;       // 0..255
  const int lane = tid & 31;          // wave32 lane
  const int wave = tid >> 5;          // 0..7
  const int half = lane >> 4;         // half-wave select (A/B K pairing)
  const int l16  = lane & 15;

  // ---- stream row (re & im) into LDS ----
#if USE_ASYNC_LDS
  {
    // Builtin wants: (AS1 int4* gsrc, AS3 int4* ldsdst, imm offset, imm cpol)
    // (HIP prints AS1 as "__device__"). Untyped 16B copies: casts are safe.
    typedef __attribute__((address_space(1))) v4i gv4i;
    typedef __attribute__((address_space(3))) v4i lv4i;
    gv4i* sre = (gv4i*)(xre + (size_t)row * NSTATE);
    gv4i* sim = (gv4i*)(xim + (size_t)row * NSTATE);
    lv4i* dre = (lv4i*)&lds[0][0];
    lv4i* dim = (lv4i*)&lds[1][0];
    #pragma unroll
    for (int i = 0; i < (NSTATE / 4) / 256; ++i) {
      const int e = tid + 256 * i;
      __builtin_amdgcn_global_load_async_to_lds_b128(sre + e, dre + e, 0, 0);
      __builtin_amdgcn_global_load_async_to_lds_b128(sim + e, dim + e, 0, 0);
    }
#if __has_builtin(__builtin_amdgcn_s_wait_asynccnt)
    __builtin_amdgcn_s_wait_asynccnt(0);
#else
    asm volatile("s_wait_asynccnt 0x0" ::: "memory");
#endif
  }
#else
  {
    const v4f* sre = (const v4f*)(xre + (size_t)row * NSTATE);
    const v4f* sim = (const v4f*)(xim + (size_t)row * NSTATE);
    v4f* dre = (v4f*)&lds[0][0];
    v4f* dim = (v4f*)&lds[1][0];
    #pragma unroll
    for (int i = 0; i < (NSTATE / 4) / 256; ++i) {
      const int e = tid + 256 * i;
      dre[e] = __builtin_nontemporal_load(sre + e);
      dim[e] = __builtin_nontemporal_load(sim + e);
    }
  }
#endif

  // ---- H16 fragments, scaled by (1/sqrt2)^4 = 0.25 per stage (1/64 total).
  // H is symmetric, so the SAME register fragment serves as the A operand
  // (lane: M=l16, K=4c+2*half,+1) and the B operand (lane: N=l16, same K)
  // of V_WMMA_F32_16X16X4_F32.
  v2f hf[4];
  #pragma unroll
  for (int c = 0; c < 4; ++c) {
    const int k0 = 4 * c + 2 * half;
    hf[c].x = hsign(l16, k0);
    hf[c].y = hsign(l16, k0 + 1);
  }

  __syncthreads();

  // ==== stage 0: contract i0 (n = 256*i2 + 16*i1 + i0). Y = X(256x16) * H.
  // 32 jobs = {re,im} x 16 row-block tiles; tiles touch disjoint LDS rows,
  // so the stage is in-place with no intra-stage barrier.
  #pragma unroll
  for (int jj = 0; jj < 4; ++jj) {
    const int j = wave + 8 * jj;
    const int comp = j >> 4, t = j & 15;
    float* base = &lds[comp][0];
    v8f acc = {};
    #pragma unroll
    for (int c = 0; c < 4; ++c) {
      // A = X[16t+m][4c+2h .. +1]  (two consecutive floats -> ds_load_b64)
      v2f a = *(const v2f*)&base[16 * (16 * t + l16) + 4 * c + 2 * half];
      acc = wmma_f32_16x16x4(a, hf[c], acc);          // X * H
    }
    #pragma unroll
    for (int g = 0; g < 8; ++g)                        // D: M=g+8h, N=l16
      base[16 * (16 * t + g + 8 * half) + l16] = acc[g];
  }
  __syncthreads();

  // ==== stage 1: contract i1. For each slice i2=t: Y = H * V(16x16).
  #pragma unroll
  for (int jj = 0; jj < 4; ++jj) {
    const int j = wave + 8 * jj;
    const int comp = j >> 4, t = j & 15;
    float* base = &lds[comp][256 * t];
    v8f acc = {};
    #pragma unroll
    for (int c = 0; c < 4; ++c) {
      const int k0 = 4 * c + 2 * half;
      v2f b;                                           // B = V[k][n=l16]
      b.x = base[16 * k0 + l16];
      b.y = base[16 * (k0 + 1) + l16];
      acc = wmma_f32_16x16x4(hf[c], b, acc);           // H * V
    }
    #pragma unroll
    for (int g = 0; g < 8; ++g)                        // D at 16*a + i0
      base[16 * (g + 8 * half) + l16] = acc[g];
  }
  __syncthreads();

  // ==== stage 2: contract i2. W is 16x256; 16 column-block tiles: Y = H * W.
  #pragma unroll
  for (int jj = 0; jj < 4; ++jj) {
    const int j = wave + 8 * jj;
    const int comp = j >> 4, t = j & 15;
    float* base = &lds[comp][0];
    v8f acc = {};
    #pragma unroll
    for (int c = 0; c < 4; ++c) {
      const int k0 = 4 * c + 2 * half;
      v2f b;                                           // B = W[k][16t + l16]
      b.x = base[256 * k0       + 16 * t + l16];
      b.y = base[256 * (k0 + 1) + 16 * t + l16];
      acc = wmma_f32_16x16x4(hf[c], b, acc);           // H * W
    }
    #pragma unroll
    for (int g = 0; g < 8; ++g)                        // D at 256*a + col
      base[256 * (g + 8 * half) + 16 * t + l16] = acc[g];
  }
  __syncthreads();

  // ---- interleave re/im -> complex64, 128-bit NT stores (2 complex each)
  {
    float* o = out + (size_t)row * NSTATE * 2;
    #pragma unroll
    for (int i = 0; i < (NSTATE / 2) / 256; ++i) {
      const int e = tid + 256 * i;
      v4f v;
      v.x = lds[0][2 * e];     v.y = lds[1][2 * e];
      v.z = lds[0][2 * e + 1]; v.w = lds[1][2 * e + 1];
      __builtin_nontemporal_store(v, (v4f*)(o + 4 * e));
    }
  }
}

extern "C" void kernel_launch(void* const* d_in, const int* in_sizes, int n_in,
                              void* d_out, int out_size, void* d_ws, size_t ws_size,
                              hipStream_t stream) {
  (void)n_in; (void)d_ws; (void)ws_size; (void)out_size;
  const float* xre = (const float*)d_in[0];   // [8,512,4096] f32
  const float* xim = (const float*)d_in[1];   // [8,512,4096] f32
  float* out = (float*)d_out;                 // complex64 -> interleaved f32 pairs

  const int rows = in_sizes[0] / NSTATE;      // 8*512 = 4096 state vectors
  hipLaunchKernelGGL(hadamard4096_fwht_wmma, dim3(rows), dim3(256), 0, stream,
                     xre, xim, out);
}